// GraphDenseNet_25202868093188
// MI455X (gfx1250) — compile-verified
//
#include <hip/hip_runtime.h>
#include <hip/hip_bf16.h>

typedef __attribute__((ext_vector_type(2))) float v2f;
typedef __attribute__((ext_vector_type(8))) float v8f;

#define N_NODES  50000
#define N_EDGES  800000
#define D_IN     128
#define D_OUT    128
#define N_GRAPHS 512
#define N_CONVS  5
#define N_CLS    96
#define BN_EPS   1e-5f

// ---------------- workspace layout (float offsets) ----------------
#define AGG_OFF    0                                   // [N_NODES*128]
#define H_OFF      (AGG_OFF   + N_NODES*128)           // [N_NODES*128]
#define WR_OFF     (H_OFF     + N_NODES*128)           // [128*128]
#define WT_OFF     (WR_OFF    + 128*128)               // [128*128]
#define BR_OFF     (WT_OFF    + 128*128)               // [128]
#define SUM_OFF    (BR_OFF    + 128)                   // [128]
#define SUMSQ_OFF  (SUM_OFF   + 128)                   // [128]
#define MEAN_OFF   (SUMSQ_OFF + 128)                   // [128]
#define INV_OFF    (MEAN_OFF  + 128)                   // [128]
#define POOL_OFF   (INV_OFF   + 128)                   // [N_GRAPHS*128]

// ---------------- utility: zero a float range ----------------
__global__ void zero_f32(float* __restrict__ p, int n) {
    int i = blockIdx.x * blockDim.x + threadIdx.x;
    if (i < n) p[i] = 0.0f;
}

// ---------------- fold the 5 conv weights ----------------
__global__ void fold_weights(const float* __restrict__ W_rel,
                             const float* __restrict__ b_rel,
                             const float* __restrict__ W_root,
                             float* __restrict__ Wr, float* __restrict__ Wt,
                             float* __restrict__ br) {
    int idx = blockIdx.x * blockDim.x + threadIdx.x;
    if (idx >= D_OUT * D_IN) return;
    float s1 = 0.f, s2 = 0.f;
    for (int c = 0; c < N_CONVS; ++c) {
        s1 += W_rel[c * D_OUT * D_IN + idx];
        s2 += W_root[c * D_OUT * D_IN + idx];
    }
    Wr[idx] = s1;
    Wt[idx] = s2;
    if (idx < D_OUT) {
        float s3 = 0.f;
        for (int c = 0; c < N_CONVS; ++c) s3 += b_rel[c * D_OUT + idx];
        br[idx] = s3;
    }
}

// ---------------- edge aggregation: agg[dst] += x[src] ----------------
__global__ __launch_bounds__(128)
void edge_agg(const float* __restrict__ x, const int* __restrict__ ei,
              float* __restrict__ agg) {
    int e = blockIdx.x;               // one edge per block
    int c = threadIdx.x;              // one channel per thread (128)
    int src = ei[e];
    int dst = ei[N_EDGES + e];
    atomicAdd(&agg[(size_t)dst * D_IN + c], x[(size_t)src * D_IN + c]);
}

// ---------------- fused double GEMM via V_WMMA_F32_16X16X4_F32 ----------
// h[N,128] = agg @ Wr^T + x @ Wt^T + br
// Block: 256 threads = 8 waves; wave w computes the 16x16 tile at
// rows [blockIdx.x*16, +16) x cols [w*16, +16).  50000 = 3125*16 exactly.
__global__ __launch_bounds__(256)
void gemm_h_wmma(const float* __restrict__ agg, const float* __restrict__ x,
                 const float* __restrict__ Wr, const float* __restrict__ Wt,
                 const float* __restrict__ br, float* __restrict__ h) {
    const int wave = threadIdx.x >> 5;
    const int lane = threadIdx.x & 31;
    const int half = lane >> 4;       // 0: K 0..1 of step, 1: K 2..3
    const int m    = lane & 15;
    const int rowbase = blockIdx.x * 16;
    const int ncol    = wave * 16;    // 8 waves cover all 128 outputs

    const float* Arow  = agg + (size_t)(rowbase + m) * D_IN;
    const float* Xrow  = x   + (size_t)(rowbase + m) * D_IN;
    const float* WrRow = Wr  + (size_t)(ncol + m) * D_IN;   // B[k][n] = Wr[n][k]
    const float* WtRow = Wt  + (size_t)(ncol + m) * D_IN;

    v8f acc = {};
    #pragma unroll
    for (int k0 = 0; k0 < D_IN; k0 += 4) {
        const int ka = k0 + half * 2;
        v2f a = *(const v2f*)(Arow + ka);
        v2f b = *(const v2f*)(WrRow + ka);
        acc = __builtin_amdgcn_wmma_f32_16x16x4_f32(
            false, a, false, b, (short)0, acc, false, false);
    }
    #pragma unroll
    for (int k0 = 0; k0 < D_IN; k0 += 4) {
        const int ka = k0 + half * 2;
        v2f a = *(const v2f*)(Xrow + ka);
        v2f b = *(const v2f*)(WtRow + ka);
        acc = __builtin_amdgcn_wmma_f32_16x16x4_f32(
            false, a, false, b, (short)0, acc, false, false);
    }

    const float bias = br[ncol + m];
    // C/D layout: VGPR v -> row v + 8*half, col = m
    #pragma unroll
    for (int v = 0; v < 8; ++v) {
        int row = rowbase + v + 8 * half;
        h[(size_t)row * D_OUT + ncol + m] = acc[v] + bias;
    }
}

// ---------------- BN statistics: per-channel sum / sumsq ----------------
__global__ __launch_bounds__(128)
void bn_stats(const float* __restrict__ h, float* __restrict__ sums,
              float* __restrict__ sumsq) {
    const int c = threadIdx.x;        // channel
    const int chunk = (N_NODES + gridDim.x - 1) / gridDim.x;
    int start = blockIdx.x * chunk;
    int end   = start + chunk;
    if (end > N_NODES) end = N_NODES;
    float s = 0.f, sq = 0.f;
    for (int n = start; n < end; ++n) {
        float v = h[(size_t)n * D_OUT + c];
        s  += v;
        sq += v * v;
    }
    atomicAdd(&sums[c], s);
    atomicAdd(&sumsq[c], sq);
}

__global__ void bn_finalize(const float* __restrict__ sums,
                            const float* __restrict__ sumsq,
                            float* __restrict__ mean,
                            float* __restrict__ invstd) {
    int c = threadIdx.x;              // 128 threads
    float m   = sums[c]  * (1.0f / N_NODES);
    float var = sumsq[c] * (1.0f / N_NODES) - m * m;
    mean[c]   = m;
    invstd[c] = rsqrtf(var + BN_EPS);
}

// ---------------- BN apply + relu + segment-max pooling ----------------
// pooled zero-initialized; values are max(v,0) >= 0 so integer-ordered
// atomicMax on the IEEE bit pattern is exact.
__global__ __launch_bounds__(256)
void bn_pool(const float* __restrict__ h, const int* __restrict__ batch,
             const float* __restrict__ mean, const float* __restrict__ invstd,
             const float* __restrict__ gamma, const float* __restrict__ beta,
             float* __restrict__ pooled) {
    int idx = blockIdx.x * blockDim.x + threadIdx.x;
    if (idx >= N_NODES * D_OUT) return;
    int c = idx & (D_OUT - 1);
    int n = idx >> 7;
    float v = gamma[c] * (h[idx] - mean[c]) * invstd[c] + beta[c];
    float r = fmaxf(v, 0.0f);
    int g = batch[n];
    atomicMax((int*)&pooled[(size_t)g * D_OUT + c], __float_as_int(r));
}

// ---------------- classifier: out[512,96] = pooled @ Wc^T + bc ----------
// 512/16 * 96/16 = 32*6 = 192 tiles = 24 blocks x 8 waves (exact).
__global__ __launch_bounds__(256)
void classifier_wmma(const float* __restrict__ pooled,
                     const float* __restrict__ Wc,
                     const float* __restrict__ bc,
                     float* __restrict__ out) {
    const int wave = threadIdx.x >> 5;
    const int lane = threadIdx.x & 31;
    const int half = lane >> 4;
    const int m    = lane & 15;
    const int tile = blockIdx.x * 8 + wave;  // 0..191, no divergence
    const int mt = tile / 6;
    const int nt = tile % 6;
    const int rowbase = mt * 16;
    const int ncol    = nt * 16;

    const float* Arow = pooled + (size_t)(rowbase + m) * D_OUT;
    const float* Brow = Wc     + (size_t)(ncol + m) * D_OUT;  // B[k][n]=Wc[n][k]

    v8f acc = {};
    #pragma unroll
    for (int k0 = 0; k0 < D_OUT; k0 += 4) {
        const int ka = k0 + half * 2;
        v2f a = *(const v2f*)(Arow + ka);
        v2f b = *(const v2f*)(Brow + ka);
        acc = __builtin_amdgcn_wmma_f32_16x16x4_f32(
            false, a, false, b, (short)0, acc, false, false);
    }

    const float bias = bc[ncol + m];
    #pragma unroll
    for (int v = 0; v < 8; ++v) {
        int row = rowbase + v + 8 * half;
        out[(size_t)row * N_CLS + ncol + m] = acc[v] + bias;
    }
}

// ---------------- host-side launcher ----------------
extern "C" void kernel_launch(void* const* d_in, const int* in_sizes, int n_in,
                              void* d_out, int out_size, void* d_ws, size_t ws_size,
                              hipStream_t stream) {
    const float* x      = (const float*)d_in[0];
    const int*   ei     = (const int*)  d_in[1];
    const int*   batch  = (const int*)  d_in[2];
    // d_in[3] = i (dropout branch selector, i==0 -> identity): unused
    const float* W_rel  = (const float*)d_in[4];
    const float* b_rel  = (const float*)d_in[5];
    const float* W_root = (const float*)d_in[6];
    const float* gamma  = (const float*)d_in[7];
    const float* beta   = (const float*)d_in[8];
    const float* Wc     = (const float*)d_in[9];
    const float* bc     = (const float*)d_in[10];
    float* out = (float*)d_out;

    float* ws     = (float*)d_ws;
    float* agg    = ws + AGG_OFF;
    float* h      = ws + H_OFF;
    float* Wr     = ws + WR_OFF;
    float* Wt     = ws + WT_OFF;
    float* br     = ws + BR_OFF;
    float* sums   = ws + SUM_OFF;
    float* sumsq  = ws + SUMSQ_OFF;
    float* mean   = ws + MEAN_OFF;
    float* invstd = ws + INV_OFF;
    float* pooled = ws + POOL_OFF;

    // zero: agg, stats (sums+sumsq contiguous), pooled
    {
        int n = N_NODES * D_IN;
        zero_f32<<<(n + 255) / 256, 256, 0, stream>>>(agg, n);
        zero_f32<<<1, 256, 0, stream>>>(sums, 256);              // sums+sumsq
        n = N_GRAPHS * D_OUT;
        zero_f32<<<(n + 255) / 256, 256, 0, stream>>>(pooled, n);
    }

    // fold conv weights
    fold_weights<<<(D_OUT * D_IN + 255) / 256, 256, 0, stream>>>(
        W_rel, b_rel, W_root, Wr, Wt, br);

    // edge aggregation
    edge_agg<<<N_EDGES, 128, 0, stream>>>(x, ei, agg);

    // fused double GEMM (WMMA f32 16x16x4)
    gemm_h_wmma<<<N_NODES / 16, 256, 0, stream>>>(agg, x, Wr, Wt, br, h);

    // batch-norm statistics and normalization factors
    bn_stats<<<256, 128, 0, stream>>>(h, sums, sumsq);
    bn_finalize<<<1, 128, 0, stream>>>(sums, sumsq, mean, invstd);

    // BN apply + relu + segment-max pool
    {
        int n = N_NODES * D_OUT;
        bn_pool<<<(n + 255) / 256, 256, 0, stream>>>(
            h, batch, mean, invstd, gamma, beta, pooled);
    }

    // classifier (WMMA)
    classifier_wmma<<<24, 256, 0, stream>>>(pooled, Wc, bc, out);
}